// GTLayer_9268539425408
// MI455X (gfx1250) — compile-verified
//
#include <hip/hip_runtime.h>
#include <hip/hip_bf16.h>

typedef __attribute__((ext_vector_type(2))) float v2f;
typedef __attribute__((ext_vector_type(8))) float v8f;

#define DLAT 128
#define LDS_STRIDE 132        // padded row stride (dwords): conflict-free ds_load_b64
#define ROWS_PER_BLOCK 128    // 8 waves * 16 rows
#define WAVES_PER_BLOCK 8

__device__ __forceinline__ v8f wmma_f32(v2f a, v2f b, v8f c) {
  // D(16x16,f32) = A(16x4,f32) x B(4x16,f32) + C
  return __builtin_amdgcn_wmma_f32_16x16x4_f32(
      /*neg_a=*/false, a, /*neg_b=*/false, b,
      /*c_mod=*/(short)0, c, /*reuse_a=*/false, /*reuse_b=*/false);
}

extern "C" __global__ void __launch_bounds__(256)
zero4_kernel(float4* __restrict__ p, long n4) {
  long i = (long)blockIdx.x * blockDim.x + threadIdx.x;
  if (i < n4) p[i] = make_float4(0.f, 0.f, 0.f, 0.f);
}

// Fused: Q/K/V projections (f32 WMMA), per-row q.k dot, softmax-norm (H=1),
// resEmbeds = attNorm * V.  One wave per 16-row tile.
extern "C" __global__ void __launch_bounds__(256)
proj_att_kernel(const float* __restrict__ X,
                const float* __restrict__ Wq,
                const float* __restrict__ Wk,
                const float* __restrict__ Wv,
                float* __restrict__ resE,
                float* __restrict__ attOut,
                int nRows)
{
  extern __shared__ float lds[];
  const int t = threadIdx.x;
  const int blockRow0 = blockIdx.x * ROWS_PER_BLOCK;

  // ---- stage 128x128 X slab into padded LDS (coalesced float4 loads) ----
  for (int f = t; f < (ROWS_PER_BLOCK * DLAT / 4); f += 256) {
    int r  = f >> 5;        // 32 float4 per row
    int c4 = f & 31;
    int gRow = blockRow0 + r;
    float4 vdat = make_float4(0.f, 0.f, 0.f, 0.f);
    if (gRow < nRows)
      vdat = *(const float4*)(X + (size_t)gRow * DLAT + c4 * 4);
    *(float4*)(&lds[r * LDS_STRIDE + c4 * 4]) = vdat;
  }
  __syncthreads();

  const int wave = t >> 5;
  const int lane = t & 31;
  const int hi = lane >> 4;   // K-pair select (A) / row-pair select (B)
  const int lo = lane & 15;   // M index (A) / N index (B,C,D)

  const int row0 = blockRow0 + wave * 16;
  if (row0 >= nRows) return;  // wave-uniform: EXEC stays all-1s for WMMA

  // ---- A panel: a[s] covers K = 4s..4s+3 of rows row0..row0+15 ----
  const float* xrow = &lds[(wave * 16 + lo) * LDS_STRIDE + 2 * hi];
  v2f a[32];
#pragma unroll
  for (int s = 0; s < 32; ++s) {
    a[s][0] = xrow[4 * s];
    a[s][1] = xrow[4 * s + 1];
  }

  // ---- attention logits: att[r] accumulates row (r + 8*hi) partial dot ----
  float att[8];
#pragma unroll
  for (int r = 0; r < 8; ++r) att[r] = 0.f;

  for (int n0 = 0; n0 < 8; ++n0) {
    v8f q = {};
    v8f k = {};
    const int cb = n0 * 16 + lo;
#pragma unroll
    for (int s = 0; s < 32; ++s) {
      const int krow = 4 * s + 2 * hi;
      const float* bq = Wq + (size_t)krow * DLAT + cb;
      const float* bk = Wk + (size_t)krow * DLAT + cb;
      v2f bqf; bqf[0] = bq[0]; bqf[1] = bq[DLAT];
      v2f bkf; bkf[0] = bk[0]; bkf[1] = bk[DLAT];
      q = wmma_f32(a[s], bqf, q);
      k = wmma_f32(a[s], bkf, k);
    }
#pragma unroll
    for (int r = 0; r < 8; ++r) att[r] += q[r] * k[r];
  }

  // reduce dot across each 16-lane half (wave32), then clip/exp/normalize
#pragma unroll
  for (int r = 0; r < 8; ++r) {
    float v = att[r];
    v += __shfl_xor(v, 1);
    v += __shfl_xor(v, 2);
    v += __shfl_xor(v, 4);
    v += __shfl_xor(v, 8);
    v = fminf(10.0f, fmaxf(-10.0f, v));
    float e = __expf(v);
    att[r] = e / (e + 1e-8f);   // H == 1
  }

  // ---- V projection, scaled by attNorm, streamed to workspace ----
  for (int n0 = 0; n0 < 8; ++n0) {
    v8f vc = {};
    const int cb = n0 * 16 + lo;
#pragma unroll
    for (int s = 0; s < 32; ++s) {
      const int krow = 4 * s + 2 * hi;
      const float* bv = Wv + (size_t)krow * DLAT + cb;
      v2f bvf; bvf[0] = bv[0]; bvf[1] = bv[DLAT];
      vc = wmma_f32(a[s], bvf, vc);
    }
#pragma unroll
    for (int r = 0; r < 8; ++r) {
      int m = row0 + r + 8 * hi;
      resE[(size_t)m * DLAT + cb] = att[r] * vc[r];
    }
  }

  // attNorm output (one value per row; lanes lo==0 of each half own 8 rows)
  if (lo == 0) {
#pragma unroll
    for (int r = 0; r < 8; ++r)
      attOut[row0 + r + 8 * hi] = att[r];
  }
}

// One wave per edge: lane l handles cols 4l..4l+3 (full 512B row, coalesced).
extern "C" __global__ void __launch_bounds__(256)
spmm_kernel(const int* __restrict__ adj_row,
            const int* __restrict__ adj_col,
            const float* __restrict__ adj_val,
            const float* __restrict__ resE,
            float* __restrict__ out,
            int nEdges)
{
  long gid = (long)blockIdx.x * blockDim.x + threadIdx.x;
  int e = (int)(gid >> 5);
  int lane = threadIdx.x & 31;
  if (e >= nEdges) return;
  int r = adj_row[e];
  int c = adj_col[e];
  float v = adj_val[e];
  float4 m = *(const float4*)(resE + (size_t)c * DLAT + lane * 4);
  float* dst = out + (size_t)r * DLAT + lane * 4;
  atomicAdd(dst + 0, v * m.x);
  atomicAdd(dst + 1, v * m.y);
  atomicAdd(dst + 2, v * m.z);
  atomicAdd(dst + 3, v * m.w);
}

extern "C" void kernel_launch(void* const* d_in, const int* in_sizes, int n_in,
                              void* d_out, int out_size, void* d_ws, size_t ws_size,
                              hipStream_t stream) {
  const int*   adj_row = (const int*)d_in[0];
  const int*   adj_col = (const int*)d_in[1];
  const float* adj_val = (const float*)d_in[2];
  const float* embeds  = (const float*)d_in[3];
  const float* Wq      = (const float*)d_in[4];
  const float* Wk      = (const float*)d_in[5];
  const float* Wv      = (const float*)d_in[6];
  const int E = in_sizes[0];
  const int N = in_sizes[3] / DLAT;

  float* out    = (float*)d_out;                  // [N*D] spmm result
  float* attOut = out + (size_t)N * DLAT;         // [N]   attNorm
  float* resE   = (float*)d_ws;                   // [N*D] scratch

  // 1) zero the accumulation target
  {
    long n4 = (long)N * DLAT / 4;
    int blocks = (int)((n4 + 255) / 256);
    zero4_kernel<<<blocks, 256, 0, stream>>>((float4*)out, n4);
  }
  // 2) fused projections + attention + scaled V
  {
    int blocks = (N + ROWS_PER_BLOCK - 1) / ROWS_PER_BLOCK;
    size_t ldsBytes = (size_t)ROWS_PER_BLOCK * LDS_STRIDE * sizeof(float);
    proj_att_kernel<<<blocks, 256, ldsBytes, stream>>>(
        embeds, Wq, Wk, Wv, resE, attOut, N);
  }
  // 3) edge scatter-add (atomics resolve in the 192MB L2)
  {
    long threads = (long)E * 32;
    int blocks = (int)((threads + 255) / 256);
    spmm_kernel<<<blocks, 256, 0, stream>>>(adj_row, adj_col, adj_val, resE, out, E);
  }
}